// MANN_33105607918033
// MI455X (gfx1250) — compile-verified
//
#include <hip/hip_runtime.h>
#include <hip/hip_bf16.h>
#include <math.h>
#include <stdint.h>

// ---------------------------------------------------------------------------
// MANN on gfx1250 (wave32):
//   gating MLP -> g[512,8]
//   per layer: partial[z] = g[:,z] * (x @ Wk[z]^T)   (one expert per blockIdx.z,
//              fp32 WMMA 16x16x4, async-to-LDS double-buffered staging)
//   reduce: out = act( sum_z partial[z] + g @ bk )   (deterministic order)
// ---------------------------------------------------------------------------

typedef float v2f __attribute__((ext_vector_type(2)));
typedef float v8f __attribute__((ext_vector_type(8)));

#define B_ROWS 512
#define IN_DIM 480
#define OUT_DIM 400
#define HID 512
#define KE 8          // experts
#define GH 128        // gating hidden
#define NG 32         // gating input slice (last NG columns of x)

__device__ __forceinline__ float elu_f(float v) {
    return v > 0.0f ? v : expm1f(v);
}

// ---- gfx1250 async global->LDS copy (ASYNCcnt-tracked DMA, no VGPR transit) --
__device__ __forceinline__ void async_copy_b128(uint32_t lds_byte_addr,
                                                const float* gptr) {
    asm volatile("global_load_async_to_lds_b128 %0, %1, off"
                 :: "v"(lds_byte_addr), "v"(gptr) : "memory");
}
__device__ __forceinline__ void wait_async0() {
    asm volatile("s_wait_asynccnt 0" ::: "memory");
}

// ---------------------------------------------------------------------------
// Gating MLP: x[:, -32:] -> elu(.@GW1+b) -> elu(.@GW2+b) -> .@GW3+b = g[512,8]
// ---------------------------------------------------------------------------
__global__ __launch_bounds__(128)
void mann_gating_kernel(const float* __restrict__ x,
                        const float* __restrict__ GW1, const float* __restrict__ Gb1,
                        const float* __restrict__ GW2, const float* __restrict__ Gb2,
                        const float* __restrict__ GW3, const float* __restrict__ Gb3,
                        float* __restrict__ gOut)
{
    __shared__ float xg[4][NG];
    __shared__ float h1[4][GH];
    __shared__ float h2[4][GH];

    const int t    = threadIdx.x;          // 0..127
    const int row0 = blockIdx.x * 4;

    {
        int r = t >> 5, c = t & 31;
        xg[r][c] = x[(row0 + r) * IN_DIM + (IN_DIM - NG) + c];
    }
    __syncthreads();

    #pragma unroll
    for (int r = 0; r < 4; ++r) {
        float acc = Gb1[t];
        #pragma unroll
        for (int j = 0; j < NG; ++j) acc += xg[r][j] * GW1[j * GH + t];
        h1[r][t] = elu_f(acc);
    }
    __syncthreads();

    #pragma unroll
    for (int r = 0; r < 4; ++r) {
        float acc = Gb2[t];
        for (int j = 0; j < GH; ++j) acc += h1[r][j] * GW2[j * GH + t];
        h2[r][t] = elu_f(acc);
    }
    __syncthreads();

    if (t < 4 * KE) {
        int r = t >> 3, k = t & 7;
        float acc = Gb3[k];
        for (int j = 0; j < GH; ++j) acc += h2[r][j] * GW3[j * KE + k];
        gOut[(row0 + r) * KE + k] = acc;
    }
}

// ---------------------------------------------------------------------------
// One expert's GEMM segment:  Partial[z][b,o] = g[b,z] * sum_i x[b,i]*W[z,o,i]
// Block: 256 threads (8 waves), tile 128(m) x 64(n), K-chunk 32,
// double-buffered async-to-LDS staging, fp32 WMMA 16x16x4, 2x2 accs per wave.
// ---------------------------------------------------------------------------
template <int IN, int O>
__global__ __launch_bounds__(256)
void mann_expert_kernel(const float* __restrict__ Xin,   // [512, IN]
                        const float* __restrict__ G,     // [512, 8]
                        const float* __restrict__ W,     // [KE, O, IN]
                        float* __restrict__ Partial)     // [KE, 512, O]
{
    constexpr int BM = 128, BN = 64, KT = 32;
    constexpr int ST  = 36;            // padded row stride (dwords): 36*d != 2 mod 64
    constexpr int ASZ = BM * ST;       // dwords per A buffer
    constexpr int BSZ = BN * ST;       // dwords per B buffer

    __shared__ __align__(16) float lds[2 * ASZ + 2 * BSZ];
    __shared__ uint32_t lds_base_slot;

    const int tid   = threadIdx.x;
    const int lane  = tid & 31;
    const int wid   = tid >> 5;
    const int wm    = wid & 3;     // wave m-position: 32 rows each
    const int wn    = wid >> 2;    // wave n-position: 32 cols each
    const int tileM = blockIdx.y * BM;
    const int tileN = blockIdx.x * BN;
    const int z     = blockIdx.z;  // expert

    // Obtain the 32-bit LDS byte address of `lds` as a *runtime* value.
    // The addrspacecast(@lds) constexpr is only ever a store operand here
    // (ordinary ISel: aperture | abs32-reloc offset); the volatile reload
    // gives an instruction result, so no constexpr can reach the asm below.
    {
        volatile uint32_t* slot = &lds_base_slot;
        *slot = (uint32_t)(uintptr_t)(const void*)lds;   // all lanes, same value
    }
    const uint32_t ldsBase = *(volatile uint32_t*)&lds_base_slot;

    // stage one K-chunk (32 dwords per row) into LDS buffer `buf` via async DMA
    auto stage = [&](int buf, int c) {
        const int i0 = c * KT;
        // A tile: 128 rows x 32 dwords  -> 4 x b128 per thread
        #pragma unroll
        for (int it = 0; it < 4; ++it) {
            int idx = tid + it * 256;
            int row = idx >> 3;            // 0..127
            int col = (idx & 7) << 2;      // 0,4,..,28
            uint32_t la = ldsBase + (uint32_t)((buf * ASZ + row * ST + col) * 4);
            async_copy_b128(la, Xin + (size_t)(tileM + row) * IN + i0 + col);
        }
        // B tile (row-major [o][i]): 64 rows x 32 dwords -> 2 x b128 per thread
        #pragma unroll
        for (int it = 0; it < 2; ++it) {
            int idx = tid + it * 256;
            int row = idx >> 3;            // 0..63  (= local o)
            int col = (idx & 7) << 2;
            int ocol = tileN + row;
            if (ocol < O) {                // skip OOB rows (cols >= O never stored)
                uint32_t la = ldsBase +
                    (uint32_t)((2 * ASZ + buf * BSZ + row * ST + col) * 4);
                async_copy_b128(la, W + ((size_t)z * O + ocol) * IN + i0 + col);
            }
        }
    };

    const v8f vz = {0.f, 0.f, 0.f, 0.f, 0.f, 0.f, 0.f, 0.f};
    v8f acc[2][2] = {{vz, vz}, {vz, vz}};

    const int mrow  = lane & 15;        // A: M row / B: N col within 16
    const int khalf = (lane >> 4) << 1; // 0 or 2

    constexpr int NCH = IN / KT;        // 15 or 16 chunks (IN % 32 == 0)
    stage(0, 0);

    for (int c = 0; c < NCH; ++c) {
        wait_async0();                  // this wave's DMA for chunk c done
        __syncthreads();                // => everyone's DMA done, prev reads done
        if (c + 1 < NCH) stage((c + 1) & 1, c + 1);   // overlap next chunk's DMA

        // buffer bases as runtime GEPs (NO pointer arrays: a constant array of
        // LDS pointers becomes a static initializer with addrspacecast -> lld
        // cannot relocate it)
        const float* Ab = lds + (c & 1) * ASZ;
        const float* Bb = lds + 2 * ASZ + (c & 1) * BSZ;

        #pragma unroll
        for (int s = 0; s < 8; ++s) {   // 8 k-steps of 4
            v2f af[2], bf[2];
            #pragma unroll
            for (int mi = 0; mi < 2; ++mi) {
                const float* p = Ab + (wm * 32 + mi * 16 + mrow) * ST + s * 4 + khalf;
                v2f t = {p[0], p[1]}; af[mi] = t;
            }
            #pragma unroll
            for (int ni = 0; ni < 2; ++ni) {
                const float* p = Bb + (wn * 32 + ni * 16 + mrow) * ST + s * 4 + khalf;
                v2f t = {p[0], p[1]}; bf[ni] = t;   // B[k][n] = Bs[n][k]
            }
            #pragma unroll
            for (int mi = 0; mi < 2; ++mi)
                #pragma unroll
                for (int ni = 0; ni < 2; ++ni)
                    acc[mi][ni] = __builtin_amdgcn_wmma_f32_16x16x4_f32(
                        false, af[mi], false, bf[ni],
                        (short)0, acc[mi][ni], false, false);
        }
    }

    // epilogue: scale by g[b,z], store partial
    const int nloc  = lane & 15;
    const int mhalf = lane >> 4;        // C/D layout: VGPR j -> row mhalf*8 + j
    float* P = Partial + (size_t)z * B_ROWS * O;
    #pragma unroll
    for (int mi = 0; mi < 2; ++mi) {
        #pragma unroll
        for (int j = 0; j < 8; ++j) {
            int row = tileM + wm * 32 + mi * 16 + mhalf * 8 + j;
            float gv = G[row * KE + z];
            #pragma unroll
            for (int ni = 0; ni < 2; ++ni) {
                int col = tileN + wn * 32 + ni * 16 + nloc;
                if (col < O) P[(size_t)row * O + col] = gv * acc[mi][ni][j];
            }
        }
    }
}

// ---------------------------------------------------------------------------
// Deterministic reduce: out = act( sum_z partial[z] + g @ bk )
// ---------------------------------------------------------------------------
template <int O, bool ACT>
__global__ __launch_bounds__(256)
void mann_reduce_kernel(const float* __restrict__ Partial,  // [KE, 512, O]
                        const float* __restrict__ G,        // [512, 8]
                        const float* __restrict__ bk,       // [KE, O]
                        float* __restrict__ Out)            // [512, O]
{
    int idx = blockIdx.x * 256 + threadIdx.x;
    if (idx >= B_ROWS * O) return;
    int row = idx / O;
    int col = idx - row * O;

    float v = 0.0f;
    #pragma unroll
    for (int zz = 0; zz < KE; ++zz) v += Partial[(size_t)zz * B_ROWS * O + idx];

    const float* gr = G + row * KE;
    #pragma unroll
    for (int k = 0; k < KE; ++k) v += gr[k] * bk[k * O + col];

    if (ACT) v = elu_f(v);
    Out[idx] = v;
}

// ---------------------------------------------------------------------------
extern "C" void kernel_launch(void* const* d_in, const int* in_sizes, int n_in,
                              void* d_out, int out_size, void* d_ws, size_t ws_size,
                              hipStream_t stream)
{
    (void)in_sizes; (void)n_in; (void)out_size; (void)ws_size;

    const float* x   = (const float*)d_in[0];
    // d_in[1] = gating_idx (arange: IN_DIM-NG .. IN_DIM-1) — folded in as constants
    const float* GW1 = (const float*)d_in[2];
    const float* Gb1 = (const float*)d_in[3];
    const float* GW2 = (const float*)d_in[4];
    const float* Gb2 = (const float*)d_in[5];
    const float* GW3 = (const float*)d_in[6];
    const float* Gb3 = (const float*)d_in[7];
    const float* Wk1 = (const float*)d_in[8];
    const float* bk1 = (const float*)d_in[9];
    const float* Wk2 = (const float*)d_in[10];
    const float* bk2 = (const float*)d_in[11];
    const float* Wk3 = (const float*)d_in[12];
    const float* bk3 = (const float*)d_in[13];
    float* out = (float*)d_out;

    // workspace: g[512*8] | y1[512*512] | y2[512*512] | partial[8*512*512] ~ 10 MB
    float* g       = (float*)d_ws;
    float* y1      = g  + B_ROWS * KE;
    float* y2      = y1 + B_ROWS * HID;
    float* partial = y2 + B_ROWS * HID;

    // 1) gating MLP
    mann_gating_kernel<<<B_ROWS / 4, 128, 0, stream>>>(x, GW1, Gb1, GW2, Gb2,
                                                       GW3, Gb3, g);

    // 2) layer 1: x[512,480] -> y1[512,512]
    mann_expert_kernel<IN_DIM, HID>
        <<<dim3(HID / 64, B_ROWS / 128, KE), 256, 0, stream>>>(x, g, Wk1, partial);
    mann_reduce_kernel<HID, true>
        <<<(B_ROWS * HID + 255) / 256, 256, 0, stream>>>(partial, g, bk1, y1);

    // 3) layer 2: y1 -> y2[512,512]
    mann_expert_kernel<HID, HID>
        <<<dim3(HID / 64, B_ROWS / 128, KE), 256, 0, stream>>>(y1, g, Wk2, partial);
    mann_reduce_kernel<HID, true>
        <<<(B_ROWS * HID + 255) / 256, 256, 0, stream>>>(partial, g, bk2, y2);

    // 4) layer 3: y2 -> out[512,400]
    mann_expert_kernel<HID, OUT_DIM>
        <<<dim3((OUT_DIM + 63) / 64, B_ROWS / 128, KE), 256, 0, stream>>>(y2, g, Wk3, partial);
    mann_reduce_kernel<OUT_DIM, false>
        <<<(B_ROWS * OUT_DIM + 255) / 256, 256, 0, stream>>>(partial, g, bk3, out);
}